// StructureDecoder_6760278524060
// MI455X (gfx1250) — compile-verified
//
#include <hip/hip_runtime.h>
#include <math.h>

#define N_NODES 8192
#define N_EDGES 262144
#define D 64
#define TILE 128
#define LDS_STRIDE 68   // 64 + 4 pad -> row m starts at bank 4m (mod 64): conflict-free

typedef __attribute__((ext_vector_type(2))) float v2f;
typedef __attribute__((ext_vector_type(4))) float v4f;
typedef __attribute__((ext_vector_type(8))) float v8f;

// ---------------- Stage 1: XW = X @ W  (8192x64 @ 64x64, tiny) ----------------
__global__ void xw_kernel(const float* __restrict__ x, const float* __restrict__ W,
                          float* __restrict__ xw) {
    int t = blockIdx.x * blockDim.x + threadIdx.x;   // one thread per output elem
    int m = t >> 6, n = t & 63;
    const float* xr = x + (size_t)m * D;
    float acc = 0.f;
#pragma unroll 16
    for (int k = 0; k < D; ++k) acc = fmaf(xr[k], W[k * D + n], acc);
    xw[t] = acc;
}

// ---------------- Stage 2: degrees (self-loop -> init to 1) ----------------
__global__ void deg_init_kernel(float* __restrict__ deg) {
    deg[blockIdx.x * blockDim.x + threadIdx.x] = 1.0f;
}

__global__ void deg_count_kernel(const int* __restrict__ ei, float* __restrict__ deg) {
    int e = blockIdx.x * blockDim.x + threadIdx.x;
    atomicAdd(&deg[ei[N_EDGES + e]], 1.0f);          // dst row of edge_index
}

__global__ void dinv_kernel(float* __restrict__ deg) {
    int i = blockIdx.x * blockDim.x + threadIdx.x;
    deg[i] = rsqrtf(deg[i]);                          // deg >= 1 always (self-loop)
}

// ---------------- Stage 3: self-loop contribution initializes agg ----------------
__global__ void selfagg_kernel(const float* __restrict__ xw, const float* __restrict__ dinv,
                               float* __restrict__ agg) {
    int t = blockIdx.x * blockDim.x + threadIdx.x;
    int i = t >> 6;
    float di = dinv[i];
    agg[t] = xw[t] * (di * di);
}

// ---------------- Stage 4: edge scatter-add, one lane per (edge, feature) ----------------
__global__ void scatter_kernel(const float* __restrict__ xw, const float* __restrict__ dinv,
                               const int* __restrict__ ei, float* __restrict__ agg) {
    unsigned t = blockIdx.x * blockDim.x + threadIdx.x;   // N_EDGES * 64 lanes
    int e = t >> 6;
    int d = t & 63;
    int s  = ei[e];
    int dd = ei[N_EDGES + e];
    float norm = dinv[s] * dinv[dd];
    atomicAdd(&agg[(size_t)dd * D + d], xw[(size_t)s * D + d] * norm);
}

// ---------------- Stage 5: h = relu(agg + b), in place ----------------
__global__ void relu_bias_kernel(float* __restrict__ h, const float* __restrict__ b) {
    int t = blockIdx.x * blockDim.x + threadIdx.x;
    float v = h[t] + b[t & 63];
    h[t] = v > 0.f ? v : 0.f;
}

// ---------------- Stage 6: adj = h @ h^T via V_WMMA_F32_16X16X4_F32 ----------------
// Block = 256 threads (8 waves) computes a 128x128 tile of adj.
// K = 64 fits entirely in LDS: two 128x64 fp32 strips of h (padded stride 68).
__global__ __launch_bounds__(256) void adj_wmma_kernel(const float* __restrict__ h,
                                                       float* __restrict__ out) {
    __shared__ float ldsA[TILE * LDS_STRIDE];
    __shared__ float ldsB[TILE * LDS_STRIDE];

    const int rowBase = blockIdx.y * TILE;
    const int colBase = blockIdx.x * TILE;
    const int t = threadIdx.x;

    // Cooperative load: 128 rows x 64 cols per strip; float4 per slot, coalesced.
#pragma unroll
    for (int i = 0; i < 8; ++i) {
        int idx = t + i * 256;            // 0..2047 float4 slots
        int row = idx >> 4;
        int c4  = (idx & 15) << 2;
        v4f a = *(const v4f*)(h + (size_t)(rowBase + row) * D + c4);
        v4f b = *(const v4f*)(h + (size_t)(colBase + row) * D + c4);
        *(v4f*)(&ldsA[row * LDS_STRIDE + c4]) = a;
        *(v4f*)(&ldsB[row * LDS_STRIDE + c4]) = b;
    }
    __syncthreads();

    const int wave   = t >> 5;            // 0..7 -> 16-row strip of the tile
    const int lane   = t & 31;
    const int lane16 = lane & 15;
    const int hi     = lane >> 4;         // 0/1 lane-half
    const int khalf  = hi << 1;           // K offset 0 or 2 within a 4-wide K step

    // A fragment source: row (m) fixed per lane, two consecutive K -> ds_load_b64
    const float* aRow = &ldsA[(wave * 16 + lane16) * LDS_STRIDE + khalf];

#pragma unroll 1
    for (int nsub = 0; nsub < 8; ++nsub) {
        // B = h^T: B[k][n] = h[colBase+n][k]; lane holds n=lane16, K pair at khalf
        const float* bRow = &ldsB[(nsub * 16 + lane16) * LDS_STRIDE + khalf];
        v8f c = {};
#pragma unroll
        for (int k0 = 0; k0 < D; k0 += 4) {
            v2f a = *(const v2f*)(aRow + k0);
            v2f b = *(const v2f*)(bRow + k0);
            // 8 args: (neg_a, A, neg_b, B, c_mod, C, reuse_a, reuse_b)
            c = __builtin_amdgcn_wmma_f32_16x16x4_f32(false, a, false, b,
                                                      (short)0, c, false, false);
        }
        // C/D layout: VGPR r -> M = r + 8*hi, N = lane16
        const size_t col = (size_t)(colBase + nsub * 16 + lane16);
#pragma unroll
        for (int r = 0; r < 8; ++r) {
            int mrow = rowBase + wave * 16 + (hi << 3) + r;
            out[(size_t)mrow * N_NODES + col] = c[r];
        }
    }
}

extern "C" void kernel_launch(void* const* d_in, const int* in_sizes, int n_in,
                              void* d_out, int out_size, void* d_ws, size_t ws_size,
                              hipStream_t stream) {
    const float* x  = (const float*)d_in[0];
    const int*   ei = (const int*)d_in[1];   // [2, N_EDGES] row-major
    const float* W  = (const float*)d_in[2];
    const float* b  = (const float*)d_in[3];
    float* out = (float*)d_out;

    // Workspace layout (fp32): xw[N*D] | deg[N] | agg/h[N*D]  ~= 4.03 MB
    float* ws   = (float*)d_ws;
    float* xw   = ws;
    float* deg  = xw + (size_t)N_NODES * D;
    float* agg  = deg + N_NODES;

    xw_kernel       <<<(N_NODES * D) / 256, 256, 0, stream>>>(x, W, xw);
    deg_init_kernel <<<N_NODES / 256,        256, 0, stream>>>(deg);
    deg_count_kernel<<<N_EDGES / 256,        256, 0, stream>>>(ei, deg);
    dinv_kernel     <<<N_NODES / 256,        256, 0, stream>>>(deg);
    selfagg_kernel  <<<(N_NODES * D) / 256,  256, 0, stream>>>(xw, deg, agg);
    scatter_kernel  <<<(N_EDGES * D) / 256,  256, 0, stream>>>(xw, deg, ei, agg);
    relu_bias_kernel<<<(N_NODES * D) / 256,  256, 0, stream>>>(agg, b);

    dim3 grid(N_NODES / TILE, N_NODES / TILE);   // 64 x 64 tiles of 128x128
    adj_wmma_kernel<<<grid, 256, 0, stream>>>(agg, out);
}